// fused_uvu_TP_65936337928696
// MI455X (gfx1250) — compile-verified
//
#include <hip/hip_runtime.h>
#include <cstdint>
#include <cstddef>

// ---------------------------------------------------------------------------
// Fused e3nn "uvu" tensor product, MUL=64, LS=(0,1,2), 15 paths, B=50000.
//
// Roofline: ~0.96 GB HBM traffic/launch -> ~41 us at 23.3 TB/s; only ~1 GFLOP
// of sparse-3j VALU work -> firmly bandwidth-bound. WMMA is intentionally not
// used: K<=9 / N<=5 sparse contractions with per-sample diagonal weights
// cannot fill 16x16 tiles and remove zero bytes of traffic.
//
// CDNA5 paths used: wave32 coalesced VMEM with NT temporal hints (streaming
// working set >> 192 MB L2), GLOBAL_LOAD_ASYNC_TO_LDS_B32 (ASYNCcnt) staging
// of the shared in2 row, LDS b128 broadcast reads.
// ---------------------------------------------------------------------------

typedef float v2f __attribute__((ext_vector_type(2), aligned(4)));
typedef float v4f __attribute__((ext_vector_type(4), aligned(4)));

__device__ __forceinline__ float ntl(const float* p) {
  return __builtin_nontemporal_load(p);
}
__device__ __forceinline__ void nts(float* p, float v) {
  __builtin_nontemporal_store(v, p);
}
__device__ __forceinline__ void st3(float* p, float a, float b, float c) {
  v2f v; v.x = a; v.y = b;
  __builtin_nontemporal_store(v, (v2f*)p);     // backend merges b64+b32 -> b96
  __builtin_nontemporal_store(c, p + 2);
}
__device__ __forceinline__ void st5(float* p, float a, float b, float c, float d, float e) {
  v4f v; v.x = a; v.y = b; v.z = c; v.w = d;
  __builtin_nontemporal_store(v, (v4f*)p);
  __builtin_nontemporal_store(e, p + 4);
}

__global__ __launch_bounds__(512) void tp_uvu_kernel(
    const float* __restrict__ in1,   // [B, 576]
    const float* __restrict__ in2,   // [B, 9]
    const float* __restrict__ wgt,   // [B, 960]
    float* __restrict__ out,         // [B, 3264]
    int Btot)
{
  __shared__ float sx2[8][12];       // 9 floats per sample, 16B-aligned rows

  const int u  = threadIdx.x;        // 0..63  (mul index; lane-contiguous)
  const int ty = threadIdx.y;        // 0..7   (sample within block; wave-uniform)
  const int b  = blockIdx.x * 8 + ty;
  if (b >= Btot) return;             // no block-wide barriers used -> safe

  // ---- stage in2[b,0:9] into LDS via CDNA5 async DMA. Both waves of a
  // ty-group issue (duplicate identical-byte LDS writes are benign), so each
  // wave only waits on its own ASYNCcnt and no __syncthreads is needed. ------
  const unsigned lane = threadIdx.x & 31u;
  {
    const float* g2 = in2 + (size_t)b * 9;
    if (lane < 9u) {
      unsigned ldsAddr = (unsigned)(uintptr_t)(&sx2[ty][lane]); // low 32 bits = LDS offset
      unsigned goff    = lane * 4u;
      asm volatile("global_load_async_to_lds_b32 %0, %1, %2"
                   :: "v"(ldsAddr), "v"(goff), "s"(g2)
                   : "memory");
    }
  }

  // ---- overlap: issue x1 and weight loads while the DMA is in flight --------
  const float* p1 = in1 + (size_t)b * 576;
  float a0 = ntl(p1 + u);                                  // l=0 block
  float a1[3], a2[5];
  {
    const float* q = p1 + 64 + u * 3;                      // l=1 block
    v2f t = __builtin_nontemporal_load((const v2f*)q);
    a1[0] = t.x; a1[1] = t.y; a1[2] = ntl(q + 2);
  }
  {
    const float* q = p1 + 256 + u * 5;                     // l=2 block
    v4f t = __builtin_nontemporal_load((const v4f*)q);
    a2[0] = t.x; a2[1] = t.y; a2[2] = t.z; a2[3] = t.w; a2[4] = ntl(q + 4);
  }
  float wv[15];
  {
    const float* pw = wgt + (size_t)b * 960 + u;
#pragma unroll
    for (int n = 0; n < 15; ++n) wv[n] = ntl(pw + n * 64);
  }

  // ---- wait for the async DMA, then broadcast-read in2 from LDS -------------
  asm volatile("s_wait_asynccnt 0" ::: "memory");
  float s0, s1[3], s2[5];
  {
    const float* r = &sx2[ty][0];
    v4f p = *(const v4f*)(r);       // ds_load_b128 (broadcast, no conflicts)
    v4f q = *(const v4f*)(r + 4);
    s0 = p.x; s1[0] = p.y; s1[1] = p.z; s1[2] = p.w;
    s2[0] = q.x; s2[1] = q.y; s2[2] = q.z; s2[3] = q.w; s2[4] = r[8];
  }

  float* po = out + (size_t)b * 3264;

  // Pre-scaled coefficients (3j normalized to unit Frobenius, times sqrt(2*l3+1)).
  // l=1 basis order (y,z,x); l=2 basis order (xy,yz,z2,xz,x2-y2).

  // path 0: (0,0,0) oo=0
  nts(po + u, a0 * s0 * wv[0]);

  // path 1: (0,1,1) oo=64   (delta/sqrt3 * sqrt3 = 1)
  { float w = wv[1];
    st3(po + 64 + u * 3, a0 * s1[0] * w, a0 * s1[1] * w, a0 * s1[2] * w); }

  // path 2: (0,2,2) oo=256  (delta/sqrt5 * sqrt5 = 1)
  { float w = wv[2];
    st5(po + 256 + u * 5, a0 * s2[0] * w, a0 * s2[1] * w, a0 * s2[2] * w,
        a0 * s2[3] * w, a0 * s2[4] * w); }

  // path 3: (1,0,1) oo=576
  { float w = wv[3];
    st3(po + 576 + u * 3, a1[0] * s0 * w, a1[1] * s0 * w, a1[2] * s0 * w); }

  // path 4: (1,1,0) oo=768  coef 1/sqrt3
  nts(po + 768 + u,
      0.57735027f * (a1[0] * s1[0] + a1[1] * s1[1] + a1[2] * s1[2]) * wv[4]);

  // path 5: (1,1,1) oo=832  eps/sqrt6 * sqrt3 = 1/sqrt2 (cross product)
  { float w = 0.70710678f * wv[5];
    st3(po + 832 + u * 3,
        (a1[1] * s1[2] - a1[2] * s1[1]) * w,
        (a1[2] * s1[0] - a1[0] * s1[2]) * w,
        (a1[0] * s1[1] - a1[1] * s1[0]) * w); }

  // path 6: (1,1,2) oo=1024  coeffs {1/sqrt2, 1/sqrt6, 2/sqrt6}
  { float w = wv[6];
    float k0 = 0.70710678f * (a1[2] * s1[0] + a1[0] * s1[2]);
    float k1 = 0.70710678f * (a1[0] * s1[1] + a1[1] * s1[0]);
    float k2 = 0.81649658f * a1[1] * s1[1] - 0.40824829f * (a1[0] * s1[0] + a1[2] * s1[2]);
    float k3 = 0.70710678f * (a1[2] * s1[1] + a1[1] * s1[2]);
    float k4 = 0.70710678f * (a1[2] * s1[2] - a1[0] * s1[0]);
    st5(po + 1024 + u * 5, k0 * w, k1 * w, k2 * w, k3 * w, k4 * w); }

  // path 7: (1,2,1) oo=1344  coeffs {sqrt3/sqrt10, 1/sqrt10, 2/sqrt10}
  { float w = wv[7];
    float o0 = 0.54772256f * (a1[2] * s2[0] + a1[1] * s2[1] - a1[0] * s2[4])
             - 0.31622777f * a1[0] * s2[2];
    float o1 = 0.54772256f * (a1[0] * s2[1] + a1[2] * s2[3])
             + 0.63245553f * a1[1] * s2[2];
    float o2 = 0.54772256f * (a1[0] * s2[0] + a1[1] * s2[3] + a1[2] * s2[4])
             - 0.31622777f * a1[2] * s2[2];
    st3(po + 1344 + u * 3, o0 * w, o1 * w, o2 * w); }

  // path 8: (1,2,2) oo=1536  (angular-momentum generators, sign flipped vs 212)
  { float w = wv[8];
    float k0 = -(0.40824829f * (-s2[1] * a1[0] + s2[3] * a1[2]) - 0.81649658f * s2[4] * a1[1]);
    float k1 = -(0.40824829f * ( s2[0] * a1[0] - s2[3] * a1[1] + s2[4] * a1[2])
               + 0.70710678f * s2[2] * a1[2]);
    float k2 = -(0.70710678f * ( s2[3] * a1[0] - s2[1] * a1[2]));
    float k3 = -(0.40824829f * ( s2[4] * a1[0] + s2[1] * a1[1] - s2[0] * a1[2])
               - 0.70710678f * s2[2] * a1[0]);
    float k4 = -(0.40824829f * (-s2[3] * a1[0] - s2[1] * a1[2]) + 0.81649658f * s2[0] * a1[1]);
    st5(po + 1536 + u * 5, k0 * w, k1 * w, k2 * w, k3 * w, k4 * w); }

  // path 9: (2,0,2) oo=1856
  { float w = wv[9];
    st5(po + 1856 + u * 5, a2[0] * s0 * w, a2[1] * s0 * w, a2[2] * s0 * w,
        a2[3] * s0 * w, a2[4] * s0 * w); }

  // path 10: (2,1,1) oo=2176
  { float w = wv[10];
    float o0 = 0.54772256f * (a2[0] * s1[2] + a2[1] * s1[1] - a2[4] * s1[0])
             - 0.31622777f * a2[2] * s1[0];
    float o1 = 0.54772256f * (a2[1] * s1[0] + a2[3] * s1[2])
             + 0.63245553f * a2[2] * s1[1];
    float o2 = 0.54772256f * (a2[0] * s1[0] + a2[3] * s1[1] + a2[4] * s1[2])
             - 0.31622777f * a2[2] * s1[2];
    st3(po + 2176 + u * 3, o0 * w, o1 * w, o2 * w); }

  // path 11: (2,1,2) oo=2368  coeffs {1/sqrt6, sqrt3/sqrt6, 2/sqrt6}
  { float w = wv[11];
    float k0 = 0.40824829f * (-a2[1] * s1[0] + a2[3] * s1[2]) - 0.81649658f * a2[4] * s1[1];
    float k1 = 0.40824829f * ( a2[0] * s1[0] - a2[3] * s1[1] + a2[4] * s1[2])
             + 0.70710678f * a2[2] * s1[2];
    float k2 = 0.70710678f * ( a2[3] * s1[0] - a2[1] * s1[2]);
    float k3 = 0.40824829f * ( a2[4] * s1[0] + a2[1] * s1[1] - a2[0] * s1[2])
             - 0.70710678f * a2[2] * s1[0];
    float k4 = 0.40824829f * (-a2[3] * s1[0] - a2[1] * s1[2]) + 0.81649658f * a2[0] * s1[1];
    st5(po + 2368 + u * 5, k0 * w, k1 * w, k2 * w, k3 * w, k4 * w); }

  // path 12: (2,2,0) oo=2688  coef 1/sqrt5
  nts(po + 2688 + u,
      0.44721360f * (a2[0] * s2[0] + a2[1] * s2[1] + a2[2] * s2[2] +
                     a2[3] * s2[3] + a2[4] * s2[4]) * wv[12]);

  // path 13: (2,2,1) oo=2752  coeffs {1/sqrt10, sqrt3/sqrt10, 2/sqrt10}
  { float w = wv[13];
    float o0 = 0.31622777f * (s2[1] * a2[0] - s2[0] * a2[1] - s2[4] * a2[3] + s2[3] * a2[4])
             + 0.54772256f * (s2[2] * a2[3] - s2[3] * a2[2]);
    float o1 = 0.63245553f * (s2[4] * a2[0] - s2[0] * a2[4])
             + 0.31622777f * (s2[3] * a2[1] - s2[1] * a2[3]);
    float o2 = 0.31622777f * (-s2[3] * a2[0] + s2[0] * a2[3] - s2[4] * a2[1] + s2[1] * a2[4])
             + 0.54772256f * (s2[1] * a2[2] - s2[2] * a2[1]);
    st3(po + 2752 + u * 3, o0 * w, o1 * w, o2 * w); }

  // path 14: (2,2,2) oo=2944  coeffs {2/sqrt14, 1/sqrt14, sqrt3/sqrt14}
  { float w = wv[14];
    const float V1 = 0.53452248f, V2 = 0.26726124f, V3 = 0.46291005f;
    float o0 = -V1 * (a2[0] * s2[2] + a2[2] * s2[0]) + V3 * (a2[1] * s2[3] + a2[3] * s2[1]);
    float o1 =  V2 * (a2[1] * s2[2] + a2[2] * s2[1]) - V3 * (a2[1] * s2[4] + a2[4] * s2[1])
             +  V3 * (a2[0] * s2[3] + a2[3] * s2[0]);
    float o2 =  V1 * (a2[2] * s2[2] - a2[0] * s2[0] - a2[4] * s2[4])
             +  V2 * (a2[1] * s2[1] + a2[3] * s2[3]);
    float o3 =  V2 * (a2[3] * s2[2] + a2[2] * s2[3]) + V3 * (a2[3] * s2[4] + a2[4] * s2[3])
             +  V3 * (a2[0] * s2[1] + a2[1] * s2[0]);
    float o4 = -V1 * (a2[2] * s2[4] + a2[4] * s2[2]) - V3 * a2[1] * s2[1] + V3 * a2[3] * s2[3];
    st5(po + 2944 + u * 5, o0 * w, o1 * w, o2 * w, o3 * w, o4 * w); }
}

extern "C" void kernel_launch(void* const* d_in, const int* in_sizes, int n_in,
                              void* d_out, int out_size, void* d_ws, size_t ws_size,
                              hipStream_t stream) {
  (void)n_in; (void)out_size; (void)d_ws; (void)ws_size;
  const float* in1 = (const float*)d_in[0];
  const float* in2 = (const float*)d_in[1];
  const float* wgt = (const float*)d_in[2];
  float* out = (float*)d_out;

  const int Btot = in_sizes[0] / 576;          // B = 50000
  dim3 block(64, 8, 1);                        // u-lane coalesced, 8 samples/block
  dim3 grid((unsigned)((Btot + 7) / 8), 1, 1); // 6250 blocks
  hipLaunchKernelGGL(tp_uvu_kernel, grid, block, 0, stream, in1, in2, wgt, out, Btot);
}